// AttentionBasedModulator_83193516523605
// MI455X (gfx1250) — compile-verified
//
#include <hip/hip_runtime.h>
#include <stdint.h>

// Problem constants from the reference.
#define CC 384
#define NN 32
#define HW 1024   // 32*32
#define BB 8

typedef float v4f __attribute__((ext_vector_type(4)));
typedef int   v4i __attribute__((ext_vector_type(4)));

#define AS1 __attribute__((address_space(1)))
#define AS3 __attribute__((address_space(3)))

// out[b,n,c,hw] = q[b,c,hw] * (support[c,n] * dw[c])
// One block per (b,c): stage q[b,c,0:1024] (4KB) into LDS via the CDNA5 async
// global->LDS path, then each thread streams 32 non-temporal 128-bit stores.
__global__ __launch_bounds__(256) void AttentionBasedModulator_kernel(
    const float* __restrict__ sf,   // C x N
    const float* __restrict__ q,    // B x C x HW
    const float* __restrict__ dw,   // C
    float*       __restrict__ out)  // B x N x C x HW
{
    __shared__ __align__(16) float lds_q[HW];
    __shared__ float lds_s[NN];

    const int t  = threadIdx.x;       // 0..255
    const int bc = blockIdx.x;        // b*CC + c
    const int c  = bc % CC;
    const int b  = bc / CC;

    // ---- async copy of this block's query tile into LDS (16B per lane) ----
    const float* gq = q + (size_t)bc * HW + (size_t)t * 4;
#if defined(__gfx1250__) && __has_builtin(__builtin_amdgcn_global_load_async_to_lds_b128)
    __builtin_amdgcn_global_load_async_to_lds_b128(
        (AS1 v4i*)gq, (AS3 v4i*)&lds_q[t * 4], /*offset=*/0, /*cpol=*/0);
#elif defined(__gfx1250__)
    {
        // Low 32 bits of a generic LDS pointer == LDS byte offset (aperture rule).
        unsigned lds_off = (unsigned)(uintptr_t)&lds_q[t * 4];
        asm volatile("global_load_async_to_lds_b128 %0, %1, off"
                     :: "v"(lds_off), "v"(gq) : "memory");
    }
#else
    ((v4f*)lds_q)[t] = *(const v4f*)gq;
#endif

    // ---- per-channel modulation scalars: s[n] = support[c,n] * dw[c] ----
    if (t < NN) {
        lds_s[t] = sf[c * NN + t] * dw[c];
    }

#if defined(__gfx1250__) && __has_builtin(__builtin_amdgcn_s_wait_asynccnt)
    __builtin_amdgcn_s_wait_asynccnt(0);
#elif defined(__gfx1250__)
    asm volatile("s_wait_asynccnt 0x0" ::: "memory");
#endif
    __syncthreads();

    // Each thread keeps its 4 query floats in registers for all 32 n.
    const v4f qv = ((const v4f*)lds_q)[t];

    float* ob = out + ((size_t)b * NN * CC + (size_t)c) * HW + (size_t)t * 4;

    // 402 MB output >> 192 MB L2: stream it with non-temporal stores so the
    // 12.6 MB query working set is not evicted.
#pragma unroll 4
    for (int n = 0; n < NN; ++n) {
        const float sn = lds_s[n];          // LDS broadcast read (conflict-free)
        v4f v = qv * sn;
        __builtin_nontemporal_store(v, (v4f*)(ob + (size_t)n * (CC * HW)));
    }
}

extern "C" void kernel_launch(void* const* d_in, const int* in_sizes, int n_in,
                              void* d_out, int out_size, void* d_ws, size_t ws_size,
                              hipStream_t stream) {
    const float* sf = (const float*)d_in[0];   // support_fibers: C x N
    const float* q  = (const float*)d_in[1];   // query_features: B x C x H x W
    const float* dw = (const float*)d_in[2];   // diag_weight:    C
    float* out = (float*)d_out;                // B x N x C x H x W

    dim3 grid(BB * CC);   // 3072 blocks
    dim3 block(256);      // 8 wave32s; 256 * float4 == 1024 == HW
    AttentionBasedModulator_kernel<<<grid, block, 0, stream>>>(sf, q, dw, out);
}